// ModelTraj_89069031784999
// MI455X (gfx1250) — compile-verified
//
#include <hip/hip_runtime.h>
#include <math.h>

typedef float v2f __attribute__((ext_vector_type(2)));
typedef float v8f __attribute__((ext_vector_type(8)));

#define MIN_DIST    1.0f
#define MAX_DIST    5.0f
#define MU          3.0f
#define SIGMA       2.0f
#define GAIN        50.0f
#define SMOOTH_W    14.0f
#define LEN_W       0.02f
#define EPSF        1e-6f
#define IMG_W       1280.0f
#define IMG_H       720.0f

#define MAXT        8      // pose tiles kept in registers (W <= 32)
#define CHUNK       8      // point tiles (of 16 points) per wave

// ---------------------------------------------------------------------------
// Kernel 1: per-pose setup.
//  - Builds the 4-rows-per-pose affine A-matrix [R | -R*pose] (row 3 = zeros)
//    packed row-major [globalRow][4] so kernel 2 lanes can fetch their
//    V_WMMA_F32_16X16X4_F32 A-fragment (lane<16: K=0,1; lane>=16: K=2,3).
//  - Thread 0 computes the pose-only loss terms (l2 + smooth + len).
// ---------------------------------------------------------------------------
__global__ void traj_setup(const float* __restrict__ poses,
                           const float* __restrict__ quats,
                           const float* __restrict__ poses0,
                           float* __restrict__ ws_base,
                           float* __restrict__ ws_A,
                           int W, int tiles) {
    int tid = threadIdx.x;
    int totalA = tiles * 64;                 // tiles * 16 rows * 4 floats
    for (int i = tid; i < totalA; i += blockDim.x) ws_A[i] = 0.0f;
    __syncthreads();

    if (tid < W) {
        float qw = quats[4 * tid + 0], qx = quats[4 * tid + 1];
        float qy = quats[4 * tid + 2], qz = quats[4 * tid + 3];
        float inv = rsqrtf(qw * qw + qx * qx + qy * qy + qz * qz);
        qw *= inv; qx *= inv; qy *= inv; qz *= inv;
        // reference: w = q0, u = -q[1:], vert = v + 2w(u x v) + 2 u x (u x v)
        //  => R = (1-2s)I + 2w[u]x + 2 u u^T,  s = |u|^2
        float w0 = qw, ux = -qx, uy = -qy, uz = -qz;
        float s = ux * ux + uy * uy + uz * uz;
        float R[9];
        R[0] = 1.0f + 2.0f * (ux * ux - s);
        R[1] = 2.0f * (ux * uy - w0 * uz);
        R[2] = 2.0f * (ux * uz + w0 * uy);
        R[3] = 2.0f * (ux * uy + w0 * uz);
        R[4] = 1.0f + 2.0f * (uy * uy - s);
        R[5] = 2.0f * (uy * uz - w0 * ux);
        R[6] = 2.0f * (ux * uz - w0 * uy);
        R[7] = 2.0f * (uy * uz + w0 * ux);
        R[8] = 1.0f + 2.0f * (uz * uz - s);
        float px = poses[3 * tid + 0], py = poses[3 * tid + 1], pz = poses[3 * tid + 2];
        for (int c = 0; c < 3; ++c) {
            float b = -(R[3 * c + 0] * px + R[3 * c + 1] * py + R[3 * c + 2] * pz);
            int row = tid * 4 + c;
            ws_A[row * 4 + 0] = R[3 * c + 0];
            ws_A[row * 4 + 1] = R[3 * c + 1];
            ws_A[row * 4 + 2] = R[3 * c + 2];
            ws_A[row * 4 + 3] = b;
        }
        // row tid*4+3 stays zero: vert=(0,0,0) -> p=0.5 -> log-odds 0 (no-op)
    }

    if (tid == 0) {
        float dx = poses[0] - poses0[0];
        float dy = poses[1] - poses0[1];
        float dz = poses[2] - poses0[2];
        float loss_l2 = sqrtf(dx * dx + dy * dy + dz * dz);
        float len = 0.0f, len0 = 0.0f;
        for (int i = 0; i < W - 1; ++i) {
            float ax = poses[3 * (i + 1) + 0] - poses[3 * i + 0];
            float ay = poses[3 * (i + 1) + 1] - poses[3 * i + 1];
            float az = poses[3 * (i + 1) + 2] - poses[3 * i + 2];
            len += sqrtf(ax * ax + ay * ay + az * az);
            float bx = poses0[3 * (i + 1) + 0] - poses0[3 * i + 0];
            float by = poses0[3 * (i + 1) + 1] - poses0[3 * i + 1];
            float bz = poses0[3 * (i + 1) + 2] - poses0[3 * i + 2];
            len0 += sqrtf(bx * bx + by * by + bz * bz);
        }
        float angsum = 0.0f;
        for (int i = 0; i < W - 2; ++i) {
            float abx = poses[3 * i + 0] - poses[3 * (i + 1) + 0];
            float aby = poses[3 * i + 1] - poses[3 * (i + 1) + 1];
            float abz = poses[3 * i + 2] - poses[3 * (i + 1) + 2];
            float acx = poses[3 * (i + 2) + 0] - poses[3 * (i + 1) + 0];
            float acy = poses[3 * (i + 2) + 1] - poses[3 * (i + 1) + 1];
            float acz = poses[3 * (i + 2) + 2] - poses[3 * (i + 1) + 2];
            float nab = sqrtf(abx * abx + aby * aby + abz * abz);
            float nac = sqrtf(acx * acx + acy * acy + acz * acz);
            float cphi = (abx * acx + aby * acy + abz * acz) / (nab * nac + EPSF);
            cphi = fminf(fmaxf(cphi, -1.0f), 1.0f);
            angsum += acosf(cphi);
        }
        float mean_ang = angsum / (float)(W - 2);
        ws_base[0] = loss_l2 + SMOOTH_W / (mean_ang + EPSF) + LEN_W * fabsf(len - len0);
    }
}

// per-(pose,point) elementwise chain: returns log(p/(1-p))
__device__ __forceinline__ float logodds_term(float vx, float vy, float vz,
                                              const float* __restrict__ Kv) {
    float ph0 = Kv[0] * vx + Kv[1] * vy + Kv[2] * vz;
    float ph1 = Kv[3] * vx + Kv[4] * vy + Kv[5] * vz;
    float ph2 = Kv[6] * vx + Kv[7] * vy + Kv[8] * vz;
    float px = __fdividef(ph0, ph2);        // fast rcp+mul; masks only compare
    float py = __fdividef(ph1, ph2);
    bool dist = (vz > MIN_DIST) && (vz < MAX_DIST);
    bool fov  = (ph2 > 0.0f) && (px > 1.0f) && (px < IMG_W - 1.0f)
                             && (py > 1.0f) && (py < IMG_H - 1.0f);
    bool mask = dist && fov;
    float d = __builtin_amdgcn_sqrtf(vx * vx + vy * vy + vz * vz);
    float qd = (d - MU) / SIGMA;
    float p = __expf(-0.5f * qd * qd) + (mask ? GAIN : 0.0f);
    p = fminf(fmaxf(p, 0.5f), 1.0f - EPSF);
    return __logf(__fdividef(p, 1.0f - p));
}

// ---------------------------------------------------------------------------
// Kernel 2: one wave32 per CHUNK point tiles (CHUNK*16 points).
// B fragment (4x16, K=4): lane<16 holds {px,py}, lane>=16 holds {pz,1}.
// A fragments for up to MAXT pose tiles (4 poses each) are preloaded into
// registers and reused across all CHUNK point tiles.
// v_wmma_f32_16x16x4_f32 -> each lane's 8-float D = two full poses'
// (x,y,z,pad) for its point column; one xor-16 shuffle merges the halves.
// All loads on the WMMA path are branchless (clamped index + value selects)
// so EXEC stays all-ones structurally.
// ---------------------------------------------------------------------------
__global__ void __launch_bounds__(256)
traj_main(const float* __restrict__ pts,
          const float* __restrict__ Kp,
          const float* __restrict__ ws_A,
          float* __restrict__ blockSums,
          int N, int nt, int tiles) {
    __shared__ float lds[8];
    int lane = threadIdx.x & 31;
    int wave = threadIdx.x >> 5;
    int col  = lane & 15;
    int half = lane >> 4;

    float Kv[9];
#pragma unroll
    for (int i = 0; i < 9; ++i) Kv[i] = Kp[i];

    // preload A fragments (zero-padded tiles contribute exactly 0 log-odds)
    v2f areg[MAXT];
#pragma unroll
    for (int t = 0; t < MAXT; ++t) {
        if (t < tiles) {
            const float* arow = ws_A + t * 64 + col * 4 + half * 2;
            areg[t].x = arow[0];
            areg[t].y = arow[1];
        } else {
            areg[t].x = 0.0f;
            areg[t].y = 0.0f;
        }
    }

    int base = (blockIdx.x * 8 + wave) * CHUNK;
    float rsum = 0.0f;

    for (int j = 0; j < CHUNK; ++j) {
        int tile = base + j;
        if (tile >= nt) break;                  // wave-uniform
        int idx = tile * 16 + col;
        bool valid = idx < N;
        int idxc = valid ? idx : (N - 1);       // clamp: loads always issued

        // branchless B fragment: lane<16 -> {px,py}; lane>=16 -> {pz,1}
        float v0 = pts[3 * idxc + 2 * half];    // x (half 0) or z (half 1)
        float v1 = pts[3 * idxc + 1];           // y (used by half 0 only)
        v2f b;
        b.x = valid ? v0 : 0.0f;
        b.y = half ? 1.0f : (valid ? v1 : 0.0f);

        float lo = 0.0f;
#pragma unroll
        for (int t = 0; t < MAXT; ++t) {
            v8f c = {0.f, 0.f, 0.f, 0.f, 0.f, 0.f, 0.f, 0.f};
            c = __builtin_amdgcn_wmma_f32_16x16x4_f32(
                    false, areg[t], false, b, (short)0, c, false, false);
            // lane<16: poses 4t,4t+1 ; lane>=16: poses 4t+2,4t+3
            lo += logodds_term(c[0], c[1], c[2], Kv);
            lo += logodds_term(c[4], c[5], c[6], Kv);
        }
        // generic remainder if W > 32 (dead for the reference's W=32)
        for (int t = MAXT; t < tiles; ++t) {
            const float* arow = ws_A + t * 64 + col * 4 + half * 2;
            v2f a; a.x = arow[0]; a.y = arow[1];
            v8f c = {0.f, 0.f, 0.f, 0.f, 0.f, 0.f, 0.f, 0.f};
            c = __builtin_amdgcn_wmma_f32_16x16x4_f32(
                    false, a, false, b, (short)0, c, false, false);
            lo += logodds_term(c[0], c[1], c[2], Kv);
            lo += logodds_term(c[4], c[5], c[6], Kv);
        }

        // merge the two pose-halves for this point column
        lo += __shfl_xor(lo, 16);
        float r = 1.0f / (1.0f + __expf(-lo));
        // count each point once (lower half, valid lanes only) - branchless
        r = (half == 0 && valid) ? r : 0.0f;
        rsum += r;
    }

    // fixed-order wave reduction (deterministic)
    rsum += __shfl_xor(rsum, 16);
    rsum += __shfl_xor(rsum, 8);
    rsum += __shfl_xor(rsum, 4);
    rsum += __shfl_xor(rsum, 2);
    rsum += __shfl_xor(rsum, 1);

    if (lane == 0) lds[wave] = rsum;
    __syncthreads();
    if (threadIdx.x == 0) {
        float s = 0.0f;
        for (int i = 0; i < 8; ++i) s += lds[i];   // fixed order
        blockSums[blockIdx.x] = s;
    }
}

// ---------------------------------------------------------------------------
// Kernel 3: deterministic fixed-order reduction of block partials + finalize.
// ---------------------------------------------------------------------------
__global__ void traj_final(const float* __restrict__ bs, int nb,
                           const float* __restrict__ ws_base,
                           float* __restrict__ out, int N) {
    __shared__ float l[256];
    float s = 0.0f;
    for (int i = threadIdx.x; i < nb; i += 256) s += bs[i];  // fixed stride order
    l[threadIdx.x] = s;
    __syncthreads();
    for (int m = 128; m > 0; m >>= 1) {
        if (threadIdx.x < (unsigned)m) l[threadIdx.x] += l[threadIdx.x + m];
        __syncthreads();
    }
    if (threadIdx.x == 0) {
        // loss_vis = 1/mean(rewards) = N / sum(rewards)
        out[0] = (float)N / l[0] + ws_base[0];
    }
}

extern "C" void kernel_launch(void* const* d_in, const int* in_sizes, int n_in,
                              void* d_out, int out_size, void* d_ws, size_t ws_size,
                              hipStream_t stream) {
    const float* pts    = (const float*)d_in[0];
    const float* poses  = (const float*)d_in[1];
    const float* quats  = (const float*)d_in[2];
    const float* poses0 = (const float*)d_in[3];
    const float* K      = (const float*)d_in[4];
    int N = in_sizes[0] / 3;
    int W = in_sizes[1] / 3;

    int tiles = (W + 3) / 4;                    // pose tiles: 4 poses / 16 rows each
    int nt    = (N + 15) / 16;                  // point tiles
    int nb    = (nt + 8 * CHUNK - 1) / (8 * CHUNK);  // blocks: 8 waves x CHUNK tiles

    float* ws        = (float*)d_ws;
    float* ws_base   = ws;                      // [0]: pose-only loss
    float* ws_A      = ws + 16;                 // tiles*64 floats, WMMA A matrix
    float* blockSums = ws + 16 + (size_t)tiles * 64;

    traj_setup<<<1, 128, 0, stream>>>(poses, quats, poses0, ws_base, ws_A, W, tiles);
    traj_main<<<nb, 256, 0, stream>>>(pts, K, ws_A, blockSums, N, nt, tiles);
    traj_final<<<1, 256, 0, stream>>>(blockSums, nb, ws_base, (float*)d_out, N);
}